// GAT_2585570312519
// MI455X (gfx1250) — compile-verified
//
#include <hip/hip_runtime.h>
#include <math.h>

typedef float v2f __attribute__((ext_vector_type(2)));
typedef float v8f __attribute__((ext_vector_type(8)));
typedef int   v4i __attribute__((ext_vector_type(4)));

#define AS_GLOBAL __attribute__((address_space(1)))
#define AS_LDS    __attribute__((address_space(3)))

#if defined(__HIP_DEVICE_COMPILE__) && \
    __has_builtin(__builtin_amdgcn_global_load_async_to_lds_b128) && \
    __has_builtin(__builtin_amdgcn_s_wait_asynccnt)
#define GATV2_ASYNC_LDS 1
#else
#define GATV2_ASYNC_LDS 0
#endif

// ---------------------------------------------------------------------------
// WMMA f32 GEMM:  C[M,Nc] = A[M,K] @ B[K,Nc]          (K%4==0, Nc%32==0)
// Block = 256 threads (8 waves) -> 128x32 output tile.
// B panel (K x 32) staged once into LDS (async-to-LDS when available),
// padded row stride 40 words so lanes 0-15 / 16-31 hit disjoint banks.
// Each wave computes two 16x16 tiles sharing one A fragment -> two
// independent V_WMMA_F32_16X16X4_F32 per k-step.
// ---------------------------------------------------------------------------
#define GEMM_WAVES 8
#define LDS_STRIDE 40

__global__ void __launch_bounds__(256) gatv2_gemm_wmma(
    const float* __restrict__ A, const float* __restrict__ B,
    float* __restrict__ C, int M, int K, int Nc)
{
    __shared__ float bsh[256 * LDS_STRIDE];          // up to K=256 rows

    const int lane = threadIdx.x & 31;
    const int wave = threadIdx.x >> 5;
    const int col0 = blockIdx.x * 32;

    // ---- stage B[0:K, col0:col0+32] into LDS (16B chunks, all threads) ----
    const int chunks = K * 8;                        // 8 float4 per row
    for (int c = threadIdx.x; c < chunks; c += 256) {
        const int kr = c >> 3;
        const int co = (c & 7) * 4;
        const float* gsrc = B + (size_t)kr * Nc + col0 + co;
        float* lds = &bsh[kr * LDS_STRIDE + co];
#if GATV2_ASYNC_LDS
        __builtin_amdgcn_global_load_async_to_lds_b128(
            (AS_GLOBAL v4i*)(AS_GLOBAL void*)const_cast<float*>(gsrc),
            (AS_LDS v4i*)(AS_LDS void*)lds, 0, 0);
#else
        *(float4*)lds = *(const float4*)gsrc;
#endif
    }
#if GATV2_ASYNC_LDS
    __builtin_amdgcn_s_wait_asynccnt(0);
#endif
    __syncthreads();

    const int tileM = blockIdx.y * GEMM_WAVES + wave;
    const int row0  = tileM * 16;
    if (row0 < M) {                                  // wave-uniform guard
        const int half = lane >> 4;                  // K sub-pair selector
        const int l15  = lane & 15;
        const size_t arow = (size_t)(row0 + l15) * (size_t)K;

        v8f acc0 = {0.f,0.f,0.f,0.f,0.f,0.f,0.f,0.f};
        v8f acc1 = {0.f,0.f,0.f,0.f,0.f,0.f,0.f,0.f};

        for (int k = 0; k < K; k += 4) {
            const int k0 = k + 2 * half;
            v2f a = *(const v2f*)(A + arow + k0);    // A frag: row l15, K=k0,k0+1
            v2f b0, b1;                              // B frags from LDS
            b0.x = bsh[k0 * LDS_STRIDE + l15];
            b0.y = bsh[(k0 + 1) * LDS_STRIDE + l15];
            b1.x = bsh[k0 * LDS_STRIDE + 16 + l15];
            b1.y = bsh[(k0 + 1) * LDS_STRIDE + 16 + l15];
            acc0 = __builtin_amdgcn_wmma_f32_16x16x4_f32(
                       false, a, false, b0, (short)0, acc0, false, false);
            acc1 = __builtin_amdgcn_wmma_f32_16x16x4_f32(
                       false, a, false, b1, (short)0, acc1, false, false);
        }
#pragma unroll
        for (int r = 0; r < 8; ++r) {
            const int m = r + 8 * half;
            C[(size_t)(row0 + m) * Nc + col0 + l15]      = acc0[r];
            C[(size_t)(row0 + m) * Nc + col0 + 16 + l15] = acc1[r];
        }
    }
}

// ---------------------------------------------------------------------------
// Elementwise helpers
// ---------------------------------------------------------------------------
__global__ void gatv2_fill(float* __restrict__ p, float v, int n)
{
    int i = blockIdx.x * blockDim.x + threadIdx.x;
    if (i < n) p[i] = v;
}

__global__ void gatv2_fix_max(float* __restrict__ p, int n)
{
    int i = blockIdx.x * blockDim.x + threadIdx.x;
    if (i < n) {
        float v = p[i];
        if (!isfinite(v)) p[i] = 0.0f;
    }
}

__global__ void gatv2_bias_act(float* __restrict__ out,
                               const float* __restrict__ bias,
                               int n, int HC, int do_elu)
{
    int i = blockIdx.x * blockDim.x + threadIdx.x;
    if (i < n) {
        float v = out[i] + bias[i % HC];
        if (do_elu) v = (v > 0.0f) ? v : expm1f(v);
        out[i] = v;
    }
}

__device__ inline void atomicMaxF(float* addr, float val)
{
    int old = __float_as_int(*addr);
    while (__int_as_float(old) < val) {
        int prev = atomicCAS((int*)addr, old, __float_as_int(val));
        if (prev == old) break;
        old = prev;
    }
}

__device__ inline float lrelu02(float v) { return (v > 0.0f) ? v : 0.2f * v; }

// ---------------------------------------------------------------------------
// Edge pass 1: logits + per-(dst,head) max.  One wave32 per edge; each lane
// owns PER contiguous channels (all in one head, since PER | C).  Per-head
// reduction = shfl_xor tree over the C/PER lanes of that head.
// ---------------------------------------------------------------------------
template<int PER>
__global__ void gatv2_edge_score(const float* __restrict__ xl,
                                 const float* __restrict__ xr,
                                 const float* __restrict__ att,
                                 const int* __restrict__ src,
                                 const int* __restrict__ dst,
                                 float* __restrict__ score,
                                 float* __restrict__ mbuf,
                                 int E, int H, int C)
{
    const int lane = threadIdx.x & 31;
    const int wid  = (blockIdx.x * blockDim.x + threadIdx.x) >> 5;
    const int nw   = (gridDim.x * blockDim.x) >> 5;
    const int HC   = H * C;
    const int t0   = lane * PER;
    const int g    = C / PER;                 // lanes per head (pow2)
    const int h0   = t0 / C;                  // this lane's head

    float attl[PER];                          // wave-invariant, hoisted
#pragma unroll
    for (int i = 0; i < PER; ++i) attl[i] = att[t0 + i];

    for (int e = wid; e < E; e += nw) {
        const int s = src[e];
        const int d = dst[e];
        const float* pl = xl + (size_t)s * HC + t0;
        const float* pr = xr + (size_t)d * HC + t0;

        float dot = 0.0f;
        if (PER == 8) {                       // vector path (layers 1-2)
            const float4* l4 = (const float4*)pl;
            const float4* r4 = (const float4*)pr;
#pragma unroll
            for (int i = 0; i < 2; ++i) {
                float4 a4 = l4[i];
                float4 b4 = r4[i];
                dot += lrelu02(a4.x + b4.x) * attl[4*i + 0];
                dot += lrelu02(a4.y + b4.y) * attl[4*i + 1];
                dot += lrelu02(a4.z + b4.z) * attl[4*i + 2];
                dot += lrelu02(a4.w + b4.w) * attl[4*i + 3];
            }
        } else {
#pragma unroll
            for (int i = 0; i < PER; ++i)
                dot += lrelu02(pl[i] + pr[i]) * attl[i];
        }

        for (int off = g >> 1; off >= 1; off >>= 1)
            dot += __shfl_xor(dot, off, 32);

        if ((lane & (g - 1)) == 0) {
            score[(size_t)e * H + h0] = dot;
            atomicMaxF(&mbuf[(size_t)d * H + h0], dot);
        }
    }
}

// ---------------------------------------------------------------------------
// Edge pass 2: p = exp(score - m[dst]); denom[dst] += p
// ---------------------------------------------------------------------------
__global__ void gatv2_edge_exp(const float* __restrict__ mbuf,
                               const int* __restrict__ dst,
                               float* __restrict__ score,
                               float* __restrict__ denom,
                               int EH, int H)
{
    int i = blockIdx.x * blockDim.x + threadIdx.x;
    if (i < EH) {
        const int e = i / H;
        const int h = i - e * H;
        const int d = dst[e];
        const float p = expf(score[i] - mbuf[(size_t)d * H + h]);
        score[i] = p;
        atomicAdd(&denom[(size_t)d * H + h], p);
    }
}

// ---------------------------------------------------------------------------
// Edge pass 3: out[dst] += alpha * xl[src].  One wave32 per edge.
// ---------------------------------------------------------------------------
template<int PER>
__global__ void gatv2_edge_aggr(const float* __restrict__ xl,
                                const float* __restrict__ p,
                                const float* __restrict__ denom,
                                const int* __restrict__ src,
                                const int* __restrict__ dst,
                                float* __restrict__ out,
                                int E, int H, int C)
{
    const int lane = threadIdx.x & 31;
    const int wid  = (blockIdx.x * blockDim.x + threadIdx.x) >> 5;
    const int nw   = (gridDim.x * blockDim.x) >> 5;
    const int HC   = H * C;
    const int t0   = lane * PER;
    const int h0   = t0 / C;

    for (int e = wid; e < E; e += nw) {
        const int s = src[e];
        const int d = dst[e];
        const float alpha = p[(size_t)e * H + h0] /
                            (denom[(size_t)d * H + h0] + 1e-16f);
        const float* pl = xl + (size_t)s * HC + t0;
        float* po = out + (size_t)d * HC + t0;

        if (PER == 8) {
            const float4* l4 = (const float4*)pl;
            float4 a0 = l4[0];
            float4 a1 = l4[1];
            atomicAdd(po + 0, alpha * a0.x);
            atomicAdd(po + 1, alpha * a0.y);
            atomicAdd(po + 2, alpha * a0.z);
            atomicAdd(po + 3, alpha * a0.w);
            atomicAdd(po + 4, alpha * a1.x);
            atomicAdd(po + 5, alpha * a1.y);
            atomicAdd(po + 6, alpha * a1.z);
            atomicAdd(po + 7, alpha * a1.w);
        } else {
#pragma unroll
            for (int i = 0; i < PER; ++i)
                atomicAdd(po + i, alpha * pl[i]);
        }
    }
}

// ---------------------------------------------------------------------------
// Host-side orchestration
// ---------------------------------------------------------------------------
static void run_gemm(const float* A, const float* B, float* C,
                     int M, int K, int Nc, hipStream_t stream)
{
    dim3 block(256);                                  // 8 waves
    dim3 grid((Nc + 31) / 32,
              ((M + 15) / 16 + GEMM_WAVES - 1) / GEMM_WAVES);
    gatv2_gemm_wmma<<<grid, block, 0, stream>>>(A, B, C, M, K, Nc);
}

static void run_layer(const float* hin, int Fin,
                      const float* Wl, const float* Wr,
                      const float* att, const float* bias,
                      int H, int C, int Nn, int Ee,
                      const int* src, const int* dst,
                      float* xl, float* xr, float* score,
                      float* mbuf, float* denom,
                      float* hout, int do_elu, hipStream_t stream)
{
    const int HC = H * C;
    run_gemm(hin, Wl, xl, Nn, Fin, HC, stream);
    run_gemm(hin, Wr, xr, Nn, Fin, HC, stream);

    const int T = 256;
    gatv2_fill<<<(Nn * H + T - 1) / T, T, 0, stream>>>(mbuf, -INFINITY, Nn * H);
    gatv2_fill<<<(Nn * H + T - 1) / T, T, 0, stream>>>(denom, 0.0f, Nn * H);
    gatv2_fill<<<(Nn * HC + T - 1) / T, T, 0, stream>>>(hout, 0.0f, Nn * HC);

    const int edgeBlocks = (Ee + 7) / 8;              // 8 waves / block
    if (HC == 256) {
        gatv2_edge_score<8><<<edgeBlocks, T, 0, stream>>>(
            xl, xr, att, src, dst, score, mbuf, Ee, H, C);
    } else {
        gatv2_edge_score<2><<<edgeBlocks, T, 0, stream>>>(
            xl, xr, att, src, dst, score, mbuf, Ee, H, C);
    }
    gatv2_fix_max<<<(Nn * H + T - 1) / T, T, 0, stream>>>(mbuf, Nn * H);
    gatv2_edge_exp<<<(Ee * H + T - 1) / T, T, 0, stream>>>(mbuf, dst, score,
                                                           denom, Ee * H, H);
    if (HC == 256) {
        gatv2_edge_aggr<8><<<edgeBlocks, T, 0, stream>>>(
            xl, score, denom, src, dst, hout, Ee, H, C);
    } else {
        gatv2_edge_aggr<2><<<edgeBlocks, T, 0, stream>>>(
            xl, score, denom, src, dst, hout, Ee, H, C);
    }
    gatv2_bias_act<<<(Nn * HC + T - 1) / T, T, 0, stream>>>(hout, bias,
                                                            Nn * HC, HC, do_elu);
}

extern "C" void kernel_launch(void* const* d_in, const int* in_sizes, int n_in,
                              void* d_out, int out_size, void* d_ws, size_t ws_size,
                              hipStream_t stream)
{
    (void)n_in; (void)out_size; (void)ws_size;

    const float* x    = (const float*)d_in[0];
    const float* Wl1  = (const float*)d_in[1];
    const float* Wr1  = (const float*)d_in[2];
    const float* att1 = (const float*)d_in[3];
    const float* b1   = (const float*)d_in[4];
    const float* Wl2  = (const float*)d_in[5];
    const float* Wr2  = (const float*)d_in[6];
    const float* att2 = (const float*)d_in[7];
    const float* b2   = (const float*)d_in[8];
    const float* Wl3  = (const float*)d_in[9];
    const float* Wr3  = (const float*)d_in[10];
    const float* att3 = (const float*)d_in[11];
    const float* b3   = (const float*)d_in[12];
    const float* Wlin = (const float*)d_in[13];
    const float* blin = (const float*)d_in[14];
    const int*   ei   = (const int*)d_in[15];

    const int F_IN = 128, HEADS = 4, HID = 64, OUTC = 64;
    const int HC = HEADS * HID;                        // 256
    const int Nn = in_sizes[0] / F_IN;                 // 50000
    const int Ee = in_sizes[15] / 2;                   // 800000
    const int* src = ei;
    const int* dst = ei + Ee;

    // workspace layout (floats)
    float* xl    = (float*)d_ws;
    float* xr    = xl    + (size_t)Nn * HC;
    float* bufA  = xr    + (size_t)Nn * HC;
    float* bufB  = bufA  + (size_t)Nn * HC;
    float* bufC  = bufB  + (size_t)Nn * HC;
    float* score = bufC  + (size_t)Nn * OUTC;
    float* mbuf  = score + (size_t)Ee * HEADS;
    float* denom = mbuf  + (size_t)Nn * HEADS;

    // layer 1: [N,128] -> [N,256], ELU
    run_layer(x, F_IN, Wl1, Wr1, att1, b1, HEADS, HID, Nn, Ee,
              src, dst, xl, xr, score, mbuf, denom, bufA, 1, stream);
    // layer 2: [N,256] -> [N,256], ELU
    run_layer(bufA, HC, Wl2, Wr2, att2, b2, HEADS, HID, Nn, Ee,
              src, dst, xl, xr, score, mbuf, denom, bufB, 1, stream);
    // layer 3: [N,256] -> [N,64], single head, no activation
    run_layer(bufB, HC, Wl3, Wr3, att3, b3, 1, OUTC, Nn, Ee,
              src, dst, xl, xr, score, mbuf, denom, bufC, 0, stream);

    // final linear: [N,64] @ [64,64] + blin
    run_gemm(bufC, Wlin, (float*)d_out, Nn, OUTC, HID, stream);
    const int T = 256;
    gatv2_bias_act<<<(Nn * HID + T - 1) / T, T, 0, stream>>>(
        (float*)d_out, blin, Nn * HID, HID, 0);
}